// PoleAgent002_42125039239719
// MI455X (gfx1250) — compile-verified
//
#include <hip/hip_runtime.h>
#include <hip/hip_bf16.h>

// LRU-style linear RNN:
//   Bu = U @ B^T          (GEMM1, fp32 WMMA 16x16x4)
//   x0 = y0 @ W_y2x^T     (folded into scan kernel, scalar fp32)
//   scan: x_{t+1} = Lambda (*) x_t + Bu_t   (diagonal complex, in-place over Bu)
//   Y  = X @ W_x2y^T + b  (GEMM2, fp32 WMMA 16x16x4)
//
// Workspace layout: X[(T+1) * BATCH * NH] floats (~257 MiB).
//   GEMM1 writes Bu[t] into slice t+1; the scan overwrites it in place with
//   X[t+1]; slice 0 receives x0. GEMM2 reads the contiguous X buffer.

#define T_STEPS 512
#define BATCH   256
#define NU      32
#define NH      512
#define NH2     256
#define NY      64

typedef __attribute__((ext_vector_type(2))) float v2f;
typedef __attribute__((ext_vector_type(8))) float v8f;

__device__ __forceinline__ v8f wmma_f32(v2f a, v2f b, v8f c) {
  // D = A(16x4) * B(4x16) + C(16x16), all fp32.
  // 8-arg form: (neg_a, A, neg_b, B, c_mod, C, reuse_a, reuse_b)
  return __builtin_amdgcn_wmma_f32_16x16x4_f32(false, a, false, b, (short)0, c,
                                               false, false);
}

// ---------------------------------------------------------------------------
// GEMM1: Bu[m, h] = sum_u U[m, u] * Bm[h, u],  m = t*BATCH + b (131072 rows)
// Each wave: 16 rows x 64 cols, K = 32 (8 chained WMMAs per accumulator).
// Xout points at X + BATCH*NH so row m lands in slice t+1.
// ---------------------------------------------------------------------------
__global__ void __launch_bounds__(128) lru_bu_gemm(const float* __restrict__ U,
                                                   const float* __restrict__ Bm,
                                                   float* __restrict__ Xout) {
  const int lane  = threadIdx.x & 31;
  const int wave  = threadIdx.x >> 5;
  const int lm    = lane & 15;        // M (for A) / N (for B,C) within tile
  const int kh    = (lane >> 4) * 2;  // K sub-offset for this half-wave: 0 or 2
  const int mbase = (blockIdx.x * 4 + wave) * 16;
  const int nbase = blockIdx.y * 64;

  const float* arow = U  + (size_t)(mbase + lm) * NU + kh;
  const float* brow = Bm + (size_t)(nbase + lm) * NU + kh;

  v8f c0 = {}, c1 = {}, c2 = {}, c3 = {};
#pragma unroll
  for (int k0 = 0; k0 < NU; k0 += 4) {
    v2f a  = *(const v2f*)(arow + k0);
    v2f b0 = *(const v2f*)(brow + k0);
    v2f b1 = *(const v2f*)(brow + 16 * NU + k0);
    v2f b2 = *(const v2f*)(brow + 32 * NU + k0);
    v2f b3 = *(const v2f*)(brow + 48 * NU + k0);
    c0 = wmma_f32(a, b0, c0);
    c1 = wmma_f32(a, b1, c1);
    c2 = wmma_f32(a, b2, c2);
    c3 = wmma_f32(a, b3, c3);
  }

  const int rbase = mbase + (lane >> 4) * 8;  // C: VGPR j = rows j and j+8
#pragma unroll
  for (int j = 0; j < 8; ++j) {
    float* orow = Xout + (size_t)(rbase + j) * NH + nbase + lm;
    orow[0]  = c0[j];
    orow[16] = c1[j];
    orow[32] = c2[j];
    orow[48] = c3[j];
  }
}

// ---------------------------------------------------------------------------
// Scan: one thread per (batch b, complex channel h). Computes x0 = y0 @ Wy2x^T
// + b_y2x, writes slice 0, then walks t = 0..T-1 reading Bu[t] and rewriting
// the same addresses with X[t+1]. Coalesced: adjacent threads = adjacent h.
// ---------------------------------------------------------------------------
__global__ void __launch_bounds__(256) lru_scan(const float* __restrict__ y0,
                                                const float* __restrict__ Wy2x,
                                                const float* __restrict__ by2x,
                                                const float* __restrict__ lmr,
                                                const float* __restrict__ lmi,
                                                float* __restrict__ X) {
  const int gid = blockIdx.x * blockDim.x + threadIdx.x;
  const int b = gid >> 8;
  const int h = gid & (NH2 - 1);
  const float lr = lmr[h];
  const float li = lmi[h];

  // x0
  const float* y0r = y0   + (size_t)b * NY;
  const float* wr  = Wy2x + (size_t)h * NY;
  const float* wi  = Wy2x + (size_t)(h + NH2) * NY;
  float xr = by2x[h];
  float xi = by2x[h + NH2];
#pragma unroll 8
  for (int y = 0; y < NY; ++y) {
    const float v = y0r[y];
    xr = fmaf(v, wr[y], xr);
    xi = fmaf(v, wi[y], xi);
  }
  X[(size_t)b * NH + h]       = xr;
  X[(size_t)b * NH + h + NH2] = xi;

  // recurrence, in place over Bu (slices 1..T)
  float* p = X + (size_t)BATCH * NH + (size_t)b * NH + h;
#pragma unroll 4
  for (int t = 0; t < T_STEPS; ++t) {
    const float bur = p[0];
    const float bui = p[NH2];
    const float nr = fmaf(lr, xr, fmaf(-li, xi, bur));
    const float ni = fmaf(li, xr, fmaf(lr, xi, bui));
    p[0]   = nr;
    p[NH2] = ni;
    xr = nr;
    xi = ni;
    p += (size_t)BATCH * NH;
  }
}

// ---------------------------------------------------------------------------
// GEMM2: Y[m, n] = sum_h X[m, h] * W[n, h] + bias[n], m over 131328 rows,
// N = 64 total -> one wave covers the full output strip (4 accumulators),
// K = 512 (128 chained WMMAs per accumulator). W (128 KB) is L2-resident.
// ---------------------------------------------------------------------------
__global__ void __launch_bounds__(128) lru_out_gemm(const float* __restrict__ X,
                                                    const float* __restrict__ W,
                                                    const float* __restrict__ bias,
                                                    float* __restrict__ Y) {
  const int lane  = threadIdx.x & 31;
  const int wave  = threadIdx.x >> 5;
  const int lm    = lane & 15;
  const int kh    = (lane >> 4) * 2;
  const int mbase = (blockIdx.x * 4 + wave) * 16;

  const float* arow = X + (size_t)(mbase + lm) * NH + kh;
  const float* brow = W + (size_t)lm * NH + kh;

  v8f c0 = {}, c1 = {}, c2 = {}, c3 = {};
#pragma unroll 4
  for (int k0 = 0; k0 < NH; k0 += 4) {
    v2f a  = *(const v2f*)(arow + k0);
    v2f b0 = *(const v2f*)(brow + k0);
    v2f b1 = *(const v2f*)(brow + 16 * NH + k0);
    v2f b2 = *(const v2f*)(brow + 32 * NH + k0);
    v2f b3 = *(const v2f*)(brow + 48 * NH + k0);
    c0 = wmma_f32(a, b0, c0);
    c1 = wmma_f32(a, b1, c1);
    c2 = wmma_f32(a, b2, c2);
    c3 = wmma_f32(a, b3, c3);
  }

  const float bs0 = bias[lm];
  const float bs1 = bias[lm + 16];
  const float bs2 = bias[lm + 32];
  const float bs3 = bias[lm + 48];

  const int rbase = mbase + (lane >> 4) * 8;
#pragma unroll
  for (int j = 0; j < 8; ++j) {
    float* orow = Y + (size_t)(rbase + j) * NY + lm;
    orow[0]  = c0[j] + bs0;
    orow[16] = c1[j] + bs1;
    orow[32] = c2[j] + bs2;
    orow[48] = c3[j] + bs3;
  }
}

// ---------------------------------------------------------------------------
extern "C" void kernel_launch(void* const* d_in, const int* in_sizes, int n_in,
                              void* d_out, int out_size, void* d_ws,
                              size_t ws_size, hipStream_t stream) {
  const float* y0   = (const float*)d_in[0];  // (256, 64)
  const float* U    = (const float*)d_in[1];  // (512, 256, 32)
  const float* lmr  = (const float*)d_in[2];  // (256,)
  const float* lmi  = (const float*)d_in[3];  // (256,)
  const float* Bm   = (const float*)d_in[4];  // (512, 32)
  const float* Wy2x = (const float*)d_in[5];  // (512, 64)
  const float* by2x = (const float*)d_in[6];  // (512,)
  const float* Wx2y = (const float*)d_in[7];  // (64, 512)
  const float* bx2y = (const float*)d_in[8];  // (64,)
  float* Y = (float*)d_out;                   // (513, 256, 64)
  float* X = (float*)d_ws;                    // (513, 256, 512) floats

  // GEMM1: Bu into X slices 1..T  (M = 512*256 rows; 4 waves/WG)
  {
    dim3 grid((T_STEPS * BATCH / 16) / 4, NH / 64, 1);
    lru_bu_gemm<<<grid, 128, 0, stream>>>(U, Bm, X + (size_t)BATCH * NH);
  }
  // Scan: x0 into slice 0, then in-place recurrence over slices 1..T
  {
    lru_scan<<<(BATCH * NH2) / 256, 256, 0, stream>>>(y0, Wy2x, by2x, lmr, lmi,
                                                      X);
  }
  // GEMM2: Y = X @ W_x2y^T + b   (M = 513*256 = 131328 rows; 4 waves/WG)
  {
    dim3 grid(((T_STEPS + 1) * BATCH / 16) / 4, 1, 1);
    lru_out_gemm<<<grid, 128, 0, stream>>>(X, Wx2y, bx2y, Y);
  }
}